// BlockSSM_83614423318550
// MI455X (gfx1250) — compile-verified
//
#include <hip/hip_runtime.h>
#include <hip/hip_bf16.h>

typedef __attribute__((ext_vector_type(2))) float v2f;
typedef __attribute__((ext_vector_type(8))) float v8f;
typedef __attribute__((ext_vector_type(4))) unsigned int u32x4;
typedef __attribute__((ext_vector_type(8))) int v8i;
typedef __attribute__((ext_vector_type(4))) int v4i;

#define NXc 256
#define NYc 64
#define NUc 64
#define NDc 32
#define Tc  256
#define Bc  1024
#define BT  16          // batch rows per workgroup
#define WXS 260         // padded LDS row stride for Wx (conflict-free frag loads)
#define XS  260         // padded LDS row stride for state
#define YBS 68          // padded ybuf stride

#define XMINc (-1.0f)
#define XMAXc ( 1.0f)
#define UMINc (-1.0f)
#define UMAXc ( 1.0f)
#define UXMINc (-1.0f)
#define UXMAXc ( 1.0f)
#define DXMINc (-1.0f)
#define DXMAXc ( 1.0f)

__device__ __forceinline__ v8f wmma_f32(v2f a, v2f b, v8f c) {
    // D = A(16x4 f32) * B(4x16 f32) + C(16x16 f32)
    return __builtin_amdgcn_wmma_f32_16x16x4_f32(
        false, a, false, b, (short)0, c, false, false);
}

#if __has_builtin(__builtin_amdgcn_tensor_load_to_lds) && __has_builtin(__builtin_amdgcn_s_wait_tensorcnt)
#define USE_TDM_PRELOAD 1
#else
#define USE_TDM_PRELOAD 0
#endif

__global__ __launch_bounds__(256)
void ssm_persistent_kernel(const float* __restrict__ x0,
                           const float* __restrict__ U,
                           const float* __restrict__ D,
                           const float* __restrict__ Wx, const float* __restrict__ bx,
                           const float* __restrict__ Wu, const float* __restrict__ bu,
                           const float* __restrict__ Wd, const float* __restrict__ bd,
                           const float* __restrict__ Wy, const float* __restrict__ by,
                           float* __restrict__ Xout, float* __restrict__ Yout,
                           float* __restrict__ stat_sums)
{
    extern __shared__ float lds[];
    float* sWx    = lds;                       // [NX][WXS]   256*260 floats (padded rows)
    float* sState = sWx + NXc * WXS;           // [2][BT][XS] double-buffered state
    float* sYbuf  = sState + 2 * BT * XS;      // [BT][YBS]   y partial reduction

    const int tid   = threadIdx.x;
    const int wave  = tid >> 5;
    const int lane  = tid & 31;
    const int nl    = lane & 15;     // M index (A-frag) / N index (B,C,D frags)
    const int khalf = lane >> 4;     // upper half-wave -> K offset +2, M rows +8
    const int koff  = khalf * 2;
    const int mrow  = khalf * 8;
    const int b0    = blockIdx.x * BT;

#if USE_TDM_PRELOAD
    // ---- Wx -> LDS via Tensor Data Mover, hardware-padded to WXS stride ----
    // pad_interval encode 7 => pad after every 256 DWORDs (one 256-float row)
    // pad_amount  encode 3 => insert 4 DWORDs  => LDS row stride 260 floats
    if (wave == 0) {
        unsigned long long ga = (unsigned long long)(size_t)Wx;
        unsigned ldsoff =
            (unsigned)(unsigned long long)(__attribute__((address_space(3))) const void*)(const void*)sWx;
        u32x4 g0 = {};
        g0[0] = 1u;                                   // count=1, user descriptor
        g0[1] = ldsoff;                               // lds_addr (bytes)
        g0[2] = (unsigned)(ga & 0xFFFFFFFFull);       // global_addr[31:0]
        g0[3] = (unsigned)((ga >> 32) & 0x1FFFFFFull) // global_addr[56:32]
              | (2u << 30);                           // type = 2 ("image")
        v8i g1 = {};
        g1[0] = (int)((2u << 16)        // data_size = 4B
              |       (1u << 20)        // pad_enable
              |       (7u << 22)        // pad_interval: 256 DWORDs
              |       (3u << 25));      // pad_amount : 4 DWORDs
        g1[1] = (int)(256u << 16);      // tensor_dim0[15:0]=256 in bits 63:48
        g1[2] = (int)(256u << 16);      // tensor_dim1[15:0]=256 in bits 95:80
        g1[3] = (int)(256u << 16);      // tile_dim0 = 256 (bits 127:112)
        g1[4] = (int)256u;              // tile_dim1 = 256 (bits 143:128), tile_dim2=0
        g1[5] = (int)256u;              // tensor_dim0_stride[31:0] = 256
        g1[6] = 0;                      // stride hi, tensor_dim1_stride lo
        g1[7] = 0;
        v4i gz = {};
#if __has_include(<hip/amd_detail/amd_gfx1250_TDM.h>)
        v8i gz8 = {};
        __builtin_amdgcn_tensor_load_to_lds(g0, g1, gz, gz, gz8, 0);
#else
        __builtin_amdgcn_tensor_load_to_lds(g0, g1, gz, gz, 0);
#endif
        __builtin_amdgcn_s_wait_tensorcnt(0);
    }
#else
    // ---- fallback: manual Wx preload into LDS (padded rows) ----
    for (int i = tid; i < NXc * NXc; i += 256) {
        int r = i >> 8, c = i & 255;
        sWx[r * WXS + c] = Wx[i];
    }
#endif

    // ---- load initial state rows b0..b0+15 into buffer 0 ----
    for (int i = tid; i < BT * NXc; i += 256) {
        int r = i >> 8, c = i & 255;
        sState[r * XS + c] = x0[(size_t)(b0 + r) * NXc + c];
    }
    __syncthreads();   // publishes TDM-written Wx + state to all waves

    const int n0 = wave * 32;                   // this wave's two 16-col tiles
    // hoisted biases (t-invariant)
    const float bu0 = bu[n0 + nl],      bu1 = bu[n0 + 16 + nl];
    const float bd0 = bd[n0 + nl],      bd1 = bd[n0 + 16 + nl];
    const float bx0 = bx[n0 + nl],      bx1 = bx[n0 + 16 + nl];
    const int   yn0 = (wave & 3) * 16;          // y tile for this wave
    const int   ykb = (wave >> 2) * 128;        // K half for y
    const float by0 = by[yn0 + nl];

    // per-lane stat partial sums (raw sums; normalized at finalize)
    float s0 = 0.f, s1 = 0.f, s2 = 0.f, s3 = 0.f, s4 = 0.f, s5 = 0.f, s6 = 0.f;

    for (int t = 0; t < Tc; ++t) {
        float* Scur = sState + (t & 1)       * (BT * XS);
        float* Snxt = sState + ((t + 1) & 1) * (BT * XS);

        const float* Ut = U + ((size_t)t * Bc + b0) * NUc;
        const float* Dt = D + ((size_t)t * Bc + b0) * NDc;

        // prefetch next step's U/D tiles (global_prefetch_b8; cacheline granular)
        if (t + 1 < Tc) {
            if (tid < 32)       __builtin_prefetch(Ut + (size_t)Bc * NUc + tid * 32, 0, 3);
            else if (tid < 48)  __builtin_prefetch(Dt + (size_t)Bc * NDc + (tid - 32) * 32, 0, 3);
        }

        // ================= fu = U[t] @ Wu^T (tiles n0, n0+16) =================
        v8f aU0 = {}, aU1 = {};
        #pragma unroll
        for (int k0 = 0; k0 < NUc; k0 += 4) {
            v2f a  = *(const v2f*)(Ut + nl * NUc + k0 + koff);
            v2f w0 = *(const v2f*)(Wu + (size_t)(n0 + nl) * NUc + k0 + koff);
            v2f w1 = *(const v2f*)(Wu + (size_t)(n0 + 16 + nl) * NUc + k0 + koff);
            aU0 = wmma_f32(a, w0, aU0);
            aU1 = wmma_f32(a, w1, aU1);
        }
        #pragma unroll
        for (int j = 0; j < 8; ++j) { aU0[j] += bu0; aU1[j] += bu1; }
        #pragma unroll
        for (int j = 0; j < 8; ++j) {
            float f0 = aU0[j], f1 = aU1[j];
            s2 += fmaxf(UMINc - f0, 0.f) + fmaxf(UMINc - f1, 0.f);          // sumin
            s3 += fmaxf(f0 - UMAXc, 0.f) + fmaxf(f1 - UMAXc, 0.f);          // sumax
            s5 += fmaxf(UXMINc - f0, 0.f) + fmaxf(f0 - UXMAXc, 0.f)
                + fmaxf(UXMINc - f1, 0.f) + fmaxf(f1 - UXMAXc, 0.f);        // dx_u
        }

        // ================= fd = D[t] @ Wd^T =================
        v8f aD0 = {}, aD1 = {};
        #pragma unroll
        for (int k0 = 0; k0 < NDc; k0 += 4) {
            v2f a  = *(const v2f*)(Dt + nl * NDc + k0 + koff);
            v2f w0 = *(const v2f*)(Wd + (size_t)(n0 + nl) * NDc + k0 + koff);
            v2f w1 = *(const v2f*)(Wd + (size_t)(n0 + 16 + nl) * NDc + k0 + koff);
            aD0 = wmma_f32(a, w0, aD0);
            aD1 = wmma_f32(a, w1, aD1);
        }
        #pragma unroll
        for (int j = 0; j < 8; ++j) { aD0[j] += bd0; aD1[j] += bd1; }
        #pragma unroll
        for (int j = 0; j < 8; ++j) {
            float f0 = aD0[j], f1 = aD1[j];
            s6 += fmaxf(DXMINc - f0, 0.f) + fmaxf(f0 - DXMAXc, 0.f)
                + fmaxf(DXMINc - f1, 0.f) + fmaxf(f1 - DXMAXc, 0.f);        // dx_d
        }

        // ============ xn = xc @ Wx^T + bx + fu + fd (accumulate on top) ============
        v8f x0a, x1a;
        #pragma unroll
        for (int j = 0; j < 8; ++j) {
            x0a[j] = aU0[j] + aD0[j] + bx0;
            x1a[j] = aU1[j] + aD1[j] + bx1;
        }
        #pragma unroll 8
        for (int k0 = 0; k0 < NXc; k0 += 4) {
            v2f a  = *(const v2f*)(Scur + nl * XS + k0 + koff);
            v2f w0 = *(const v2f*)(sWx + (n0 + nl) * WXS + k0 + koff);
            v2f w1 = *(const v2f*)(sWx + (n0 + 16 + nl) * WXS + k0 + koff);
            x0a = wmma_f32(a, w0, x0a);
            x1a = wmma_f32(a, w1, x1a);
        }

        // stats on xn, sdx, write state + X output
        #pragma unroll
        for (int j = 0; j < 8; ++j) {
            float xnA = x0a[j], xnB = x1a[j];
            s0 += fmaxf(XMINc - xnA, 0.f) + fmaxf(XMINc - xnB, 0.f);        // sxmin
            s1 += fmaxf(xnA - XMAXc, 0.f) + fmaxf(xnB - XMAXc, 0.f);        // sxmax
            float xcA = Scur[(mrow + j) * XS + n0 + nl];
            float xcB = Scur[(mrow + j) * XS + n0 + 16 + nl];
            s4 += (xnA - xcA) * (xnA - xcA) + (xnB - xcB) * (xnB - xcB);    // sdx_x
            Snxt[(mrow + j) * XS + n0 + nl]      = xnA;
            Snxt[(mrow + j) * XS + n0 + 16 + nl] = xnB;
            size_t xo = ((size_t)t * Bc + b0 + mrow + j) * NXc;
            Xout[xo + n0 + nl]      = xnA;
            Xout[xo + n0 + 16 + nl] = xnB;
        }
        __syncthreads();   // B1: xn visible in Snxt; all reads of Scur done

        // ========== y = xn @ Wy^T : 4 tiles x 2 K-halves over 8 waves ==========
        v8f ya = {};
        #pragma unroll 8
        for (int k0 = 0; k0 < 128; k0 += 4) {
            int k = ykb + k0;
            v2f a = *(const v2f*)(Snxt + nl * XS + k + koff);
            v2f w = *(const v2f*)(Wy + (size_t)(yn0 + nl) * NXc + k + koff);
            ya = wmma_f32(a, w, ya);
        }
        if (wave >= 4) {
            #pragma unroll
            for (int j = 0; j < 8; ++j)
                sYbuf[(mrow + j) * YBS + yn0 + nl] = ya[j];
        }
        __syncthreads();   // B2: partials visible
        if (wave < 4) {
            #pragma unroll
            for (int j = 0; j < 8; ++j) {
                float v = ya[j] + sYbuf[(mrow + j) * YBS + yn0 + nl] + by0;
                Yout[((size_t)t * Bc + b0 + mrow + j) * NYc + yn0 + nl] = v;
            }
        }
        // next step's B1 separates this sYbuf read from the next write
    }

    // ---- global stat reduction (once) ----
    atomicAdd(&stat_sums[0], s0);
    atomicAdd(&stat_sums[1], s1);
    atomicAdd(&stat_sums[2], s2);
    atomicAdd(&stat_sums[3], s3);
    atomicAdd(&stat_sums[4], s4);
    atomicAdd(&stat_sums[5], s5);
    atomicAdd(&stat_sums[6], s6);
}

__global__ void zero_ws_kernel(float* p) {
    if (threadIdx.x < 8) p[threadIdx.x] = 0.f;
}

__global__ void finalize_kernel(const float* __restrict__ sums, float* __restrict__ reg) {
    if (threadIdx.x == 0) {
        float s = 0.f;
        #pragma unroll
        for (int i = 0; i < 7; ++i) s += sums[i];
        const float inv = 1.0f / ((float)Tc * (float)Bc * (float)NXc);
        reg[0] = 0.2f * s * inv;
    }
}

extern "C" void kernel_launch(void* const* d_in, const int* in_sizes, int n_in,
                              void* d_out, int out_size, void* d_ws, size_t ws_size,
                              hipStream_t stream) {
    const float* x  = (const float*)d_in[0];
    const float* U  = (const float*)d_in[1];
    const float* D  = (const float*)d_in[2];
    const float* Wx = (const float*)d_in[3];
    const float* bx = (const float*)d_in[4];
    const float* Wu = (const float*)d_in[5];
    const float* bu = (const float*)d_in[6];
    const float* Wd = (const float*)d_in[7];
    const float* bd = (const float*)d_in[8];
    const float* Wy = (const float*)d_in[9];
    const float* by = (const float*)d_in[10];

    float* Xout = (float*)d_out;                               // [T,B,NX]
    float* Yout = Xout + (size_t)Tc * Bc * NXc;                // [T,B,NY]
    float* reg  = Yout + (size_t)Tc * Bc * NYc;                // scalar
    float* sums = (float*)d_ws;                                // 7 partial sums

    const size_t smem = (size_t)(NXc * WXS + 2 * BT * XS + BT * YBS) * sizeof(float);
    // Allow >64KB dynamic LDS (CDNA5 WGP has 320KB)
    (void)hipFuncSetAttribute((const void*)ssm_persistent_kernel,
                              hipFuncAttributeMaxDynamicSharedMemorySize, (int)smem);

    zero_ws_kernel<<<1, 32, 0, stream>>>(sums);
    ssm_persistent_kernel<<<Bc / BT, 256, smem, stream>>>(
        x, U, D, Wx, bx, Wu, bu, Wd, bd, Wy, by, Xout, Yout, sums);
    finalize_kernel<<<1, 32, 0, stream>>>(sums, reg);
}